// Basic_91139206021707
// MI455X (gfx1250) — compile-verified
//
#include <hip/hip_runtime.h>
#include <math.h>

typedef __attribute__((ext_vector_type(2))) float v2f;
typedef __attribute__((ext_vector_type(8))) float v8f;

#define WMMA_F32(a, b, c) \
  __builtin_amdgcn_wmma_f32_16x16x4_f32(false, (a), false, (b), (short)0, (c), false, false)

// ---------------------------------------------------------------- degrees ---
__global__ void deg_kernel(const int* __restrict__ col, float* __restrict__ deg, int E) {
  int e = blockIdx.x * blockDim.x + threadIdx.x;
  if (e < E) atomicAdd(&deg[col[e]], 1.0f);
}

__global__ void dinv_kernel(const float* __restrict__ deg, float* __restrict__ dinv, int n) {
  int i = blockIdx.x * blockDim.x + threadIdx.x;
  if (i < n) dinv[i] = rsqrtf(deg[i] + 2.0f);  // deg+SELF_LOOP_W > 0 always
}

// ------------------------------------------------------------- fp32 GEMM ----
// C[M x Nout] = A[M x K] * W[K x Nout] with V_WMMA_F32_16X16X4_F32.
// Block = 128 threads (4 waves). Each wave computes 16 rows x 64 cols.
// A-layout (16x4, MxK): lane m=lane&15, VGPR v holds K = 2*(lane>>4) + v.
// B-layout (4x16, KxN): VGPR v holds row K = 2*(lane>>4) + v, col N = lane&15.
// C-layout (16x16): VGPR v holds row M = v + 8*(lane>>4), col N = lane&15.
__global__ void gemm_wmma_f32(const float* __restrict__ A, const float* __restrict__ W,
                              float* __restrict__ C, int M, int K, int Nout) {
  const int lane = threadIdx.x & 31;
  const int wv   = threadIdx.x >> 5;
  const int m    = lane & 15;
  const int kh   = lane >> 4;
  const int r0   = blockIdx.x * 64 + wv * 16;
  const int c0   = blockIdx.y * 64;
  int rowA = r0 + m; if (rowA > M - 1) rowA = M - 1;     // clamp; never stored OOB
  const float* __restrict__ arow = A + (size_t)rowA * K;

  v8f acc0 = {}, acc1 = {}, acc2 = {}, acc3 = {};
  #pragma unroll 2
  for (int k0 = 0; k0 < K; k0 += 4) {
    const int ka = k0 + 2 * kh;
    v2f a; a.x = arow[ka]; a.y = arow[ka + 1];
    const float* __restrict__ w0 = W + (size_t)ka * Nout + c0 + m;
    const float* __restrict__ w1 = w0 + Nout;
    v2f b0; b0.x = w0[0];  b0.y = w1[0];
    v2f b1; b1.x = w0[16]; b1.y = w1[16];
    v2f b2; b2.x = w0[32]; b2.y = w1[32];
    v2f b3; b3.x = w0[48]; b3.y = w1[48];
    acc0 = WMMA_F32(a, b0, acc0);
    acc1 = WMMA_F32(a, b1, acc1);
    acc2 = WMMA_F32(a, b2, acc2);
    acc3 = WMMA_F32(a, b3, acc3);
  }
  #pragma unroll
  for (int v = 0; v < 8; ++v) {
    int row = r0 + v + 8 * kh;
    if (row < M) {
      float* crow = C + (size_t)row * Nout + c0 + m;
      crow[0]  = acc0[v];
      crow[16] = acc1[v];
      crow[32] = acc2[v];
      crow[48] = acc3[v];
    }
  }
}

// ------------------------------------------------- edge scatter-aggregate ---
// One wave per edge: agg[col] += dinv[row]*dinv[col] * h[row]
__global__ void scatter_agg(const float* __restrict__ h, const int* __restrict__ row,
                            const int* __restrict__ col, const float* __restrict__ dinv,
                            float* __restrict__ agg, int E, int F) {
  const int lane = threadIdx.x & 31;
  const int e = blockIdx.x * (blockDim.x >> 5) + (threadIdx.x >> 5);
  if (e >= E) return;
  const int r = row[e], c = col[e];
  const float nrm = dinv[r] * dinv[c];
  const float* __restrict__ hs = h + (size_t)r * F;
  float* __restrict__ ad = agg + (size_t)c * F;
  for (int f = lane; f < F; f += 32) atomicAdd(&ad[f], nrm * hs[f]);
}

// agg = [relu](agg + 2*dinv^2*hpre + bias)   (in place)
__global__ void gcn_epilogue(float* __restrict__ agg, const float* __restrict__ hpre,
                             const float* __restrict__ dinv, const float* __restrict__ bias,
                             int F, int do_relu) {
  const int rowi = blockIdx.x;
  const int f = threadIdx.x;
  const float di = dinv[rowi];
  const size_t idx = (size_t)rowi * F + f;
  float v = agg[idx] + 2.0f * di * di * hpre[idx] + bias[f];
  if (do_relu) v = fmaxf(v, 0.0f);
  agg[idx] = v;
}

// -------------------------------------------------------- link predictor ----
// Wave per 16 query edges: A[16x128] = h2[src]+h2[dst]; WMMA -> [16x64] relu;
// then 64->1 dot + sigmoid via LDS + 16-lane shuffle reduce.
__global__ void link_predict(const float* __restrict__ h2,
                             const int* __restrict__ pos, const int* __restrict__ neg,
                             int Ep, int En,
                             const float* __restrict__ Wl1, const float* __restrict__ bl1,
                             const float* __restrict__ Wl2, const float* __restrict__ bl2,
                             float* __restrict__ out) {
  __shared__ float lds[4][16][65];
  const int lane = threadIdx.x & 31;
  const int wv   = threadIdx.x >> 5;
  const int Etot = Ep + En;
  const int tile = blockIdx.x * 4 + wv;
  const int m    = lane & 15;
  const int kh   = lane >> 4;

  int e = tile * 16 + m; if (e > Etot - 1) e = Etot - 1;   // clamp, no early return
  int src, dst;
  if (e < Ep) { src = pos[e];      dst = pos[Ep + e]; }
  else        { int e2 = e - Ep;   src = neg[e2]; dst = neg[En + e2]; }
  const float* __restrict__ hs = h2 + (size_t)src * 128;
  const float* __restrict__ hd = h2 + (size_t)dst * 128;

  v8f acc0 = {}, acc1 = {}, acc2 = {}, acc3 = {};
  #pragma unroll 2
  for (int k0 = 0; k0 < 128; k0 += 4) {
    const int ka = k0 + 2 * kh;
    v2f a; a.x = hs[ka] + hd[ka]; a.y = hs[ka + 1] + hd[ka + 1];
    const float* __restrict__ w0 = Wl1 + (size_t)ka * 64 + m;
    const float* __restrict__ w1 = w0 + 64;
    v2f b0; b0.x = w0[0];  b0.y = w1[0];
    v2f b1; b1.x = w0[16]; b1.y = w1[16];
    v2f b2; b2.x = w0[32]; b2.y = w1[32];
    v2f b3; b3.x = w0[48]; b3.y = w1[48];
    acc0 = WMMA_F32(a, b0, acc0);
    acc1 = WMMA_F32(a, b1, acc1);
    acc2 = WMMA_F32(a, b2, acc2);
    acc3 = WMMA_F32(a, b3, acc3);
  }
  #pragma unroll
  for (int v = 0; v < 8; ++v) {
    const int er = v + 8 * kh;
    lds[wv][er][m +  0] = fmaxf(acc0[v] + bl1[m +  0], 0.0f);
    lds[wv][er][m + 16] = fmaxf(acc1[v] + bl1[m + 16], 0.0f);
    lds[wv][er][m + 32] = fmaxf(acc2[v] + bl1[m + 32], 0.0f);
    lds[wv][er][m + 48] = fmaxf(acc3[v] + bl1[m + 48], 0.0f);
  }
  __syncthreads();

  // lane handles edge (lane&15), n-range [kh*32, kh*32+32)
  float s = 0.0f;
  #pragma unroll
  for (int n = 0; n < 32; ++n) {
    const int nn = kh * 32 + n;
    s += lds[wv][m][nn] * Wl2[nn];
  }
  s += __shfl_xor(s, 16, 32);
  s += bl2[0];
  const float sig = 1.0f / (1.0f + __expf(-s));
  const int eo = tile * 16 + m;
  if (kh == 0 && eo < Etot) out[eo] = sig;
}

// ------------------------------------------------------------------ driver --
extern "C" void kernel_launch(void* const* d_in, const int* in_sizes, int n_in,
                              void* d_out, int out_size, void* d_ws, size_t ws_size,
                              hipStream_t stream) {
  (void)n_in; (void)out_size; (void)ws_size;
  const float* x   = (const float*)d_in[0];
  const int*   tei = (const int*)d_in[1];
  const int*   pos = (const int*)d_in[2];
  const int*   neg = (const int*)d_in[3];
  const float* W1  = (const float*)d_in[4];
  const float* b1  = (const float*)d_in[5];
  const float* W2  = (const float*)d_in[6];
  const float* b2  = (const float*)d_in[7];
  const float* Wl1 = (const float*)d_in[8];
  const float* bl1 = (const float*)d_in[9];
  const float* Wl2 = (const float*)d_in[10];
  const float* bl2 = (const float*)d_in[11];

  const int Fin = 256, H1 = 256, H2 = 128;
  const int Nn   = in_sizes[0] / Fin;
  const int E    = in_sizes[1] / 2;
  const int Ep   = in_sizes[2] / 2;
  const int En   = in_sizes[3] / 2;
  const int Etot = Ep + En;
  const int* row  = tei;       // edge_index[0] = source
  const int* colv = tei + E;   // edge_index[1] = target

  float* ws = (float*)d_ws;
  size_t o = 0;
  float* deg   = ws + o; o += (size_t)Nn;
  float* dinv  = ws + o; o += (size_t)Nn;
  float* hpre1 = ws + o; o += (size_t)Nn * H1;
  float* agg1  = ws + o; o += (size_t)Nn * H1;
  float* hpre2 = ws + o; o += (size_t)Nn * H2;
  float* agg2  = ws + o; o += (size_t)Nn * H2;

  hipMemsetAsync(deg,  0, sizeof(float) * (size_t)Nn,      stream);
  hipMemsetAsync(agg1, 0, sizeof(float) * (size_t)Nn * H1, stream);
  hipMemsetAsync(agg2, 0, sizeof(float) * (size_t)Nn * H2, stream);

  deg_kernel<<<(E + 255) / 256, 256, 0, stream>>>(colv, deg, E);
  dinv_kernel<<<(Nn + 255) / 256, 256, 0, stream>>>(deg, dinv, Nn);

  const dim3 blk(128);
  // layer 1: h1 = relu(D^-1/2 A D^-1/2 (x W1) + b1)
  gemm_wmma_f32<<<dim3((Nn + 63) / 64, H1 / 64), blk, 0, stream>>>(x, W1, hpre1, Nn, Fin, H1);
  scatter_agg<<<(E + 7) / 8, 256, 0, stream>>>(hpre1, row, colv, dinv, agg1, E, H1);
  gcn_epilogue<<<Nn, H1, 0, stream>>>(agg1, hpre1, dinv, b1, H1, 1);
  // layer 2: h2 = D^-1/2 A D^-1/2 (h1 W2) + b2
  gemm_wmma_f32<<<dim3((Nn + 63) / 64, H2 / 64), blk, 0, stream>>>(agg1, W2, hpre2, Nn, H1, H2);
  scatter_agg<<<(E + 7) / 8, 256, 0, stream>>>(hpre2, row, colv, dinv, agg2, E, H2);
  gcn_epilogue<<<Nn, H2, 0, stream>>>(agg2, hpre2, dinv, b2, H2, 0);
  // link predictor over concat(pos, neg)
  const int tiles = (Etot + 15) / 16;
  link_predict<<<(tiles + 3) / 4, 128, 0, stream>>>(agg2, pos, neg, Ep, En,
                                                    Wl1, bl1, Wl2, bl2, (float*)d_out);
}